// DynamiSE_10986526343305
// MI455X (gfx1250) — compile-verified
//
#include <hip/hip_runtime.h>

#define N_NODES 50000
#define N_FEAT 128
#define HID 64
#define N_EDGES 1600000
#define LN_EPS 1e-5f

typedef float v2f __attribute__((ext_vector_type(2)));
typedef float v8f __attribute__((ext_vector_type(8)));

// D = A(16x4 f32) * B(4x16 f32) + C(16x16 f32), wave32 WMMA
__device__ __forceinline__ v8f wmma_f32(v2f a, v2f b, v8f c) {
  return __builtin_amdgcn_wmma_f32_16x16x4_f32(false, a, false, b, (short)0, c, false, false);
}

__device__ __forceinline__ float wave_sum(float v) {
#pragma unroll
  for (int m = 16; m >= 1; m >>= 1) v += __shfl_xor(v, m, 32);
  return v;
}

// ---------------- degree / dinv ----------------

__global__ void fill_kernel(float* __restrict__ p, int n, float v) {
  int i = blockIdx.x * blockDim.x + threadIdx.x;
  if (i < n) p[i] = v;
}

__global__ void deg_kernel(const long long* __restrict__ ep,
                           const long long* __restrict__ en,
                           float* __restrict__ degp, float* __restrict__ degn) {
  int i = blockIdx.x * blockDim.x + threadIdx.x;
  if (i < N_EDGES) {
    int d = (int)ep[N_EDGES + i];
    atomicAdd(&degp[d], 1.0f);
  } else if (i < 2 * N_EDGES) {
    int e = i - N_EDGES;
    int d = (int)en[N_EDGES + e];
    atomicAdd(&degn[d], 1.0f);
  }
}

__global__ void dinv_kernel(float* __restrict__ dp, float* __restrict__ dn) {
  int i = blockIdx.x * blockDim.x + threadIdx.x;
  if (i < N_NODES) {
    dp[i] = rsqrtf(dp[i]);
    dn[i] = rsqrtf(dn[i]);
  }
}

// ---------------- LayerNorm (1 wave per node, 2 feats/lane) ----------------

__global__ void ln_kernel(const float* __restrict__ hin, float* __restrict__ hn,
                          const float* __restrict__ gamma, const float* __restrict__ beta) {
  int tid = blockIdx.x * blockDim.x + threadIdx.x;
  int node = tid >> 5;
  int lane = tid & 31;
  if (node >= N_NODES) return;
  const float2 v = *(const float2*)(hin + (size_t)node * HID + 2 * lane);
  float mu = wave_sum(v.x + v.y) * (1.0f / HID);
  float dx = v.x - mu, dy = v.y - mu;
  float var = wave_sum(dx * dx + dy * dy) * (1.0f / HID);
  float inv = rsqrtf(var + LN_EPS);
  int c = 2 * lane;
  float2 o;
  o.x = dx * inv * gamma[c] + beta[c];
  o.y = dy * inv * gamma[c + 1] + beta[c + 1];
  *(float2*)(hn + (size_t)node * HID + 2 * lane) = o;
}

// ---------------- encoder GEMM: h = x @ Wenc^T + benc  (K=128) ----------------

__global__ __launch_bounds__(256) void enc_gemm_kernel(const float* __restrict__ x,
                                                       const float* __restrict__ Wenc,
                                                       const float* __restrict__ benc,
                                                       float* __restrict__ hout) {
  __shared__ float Bs[N_FEAT][HID + 1];  // Bs[k][n] = Wenc[n][k]
  int tid = threadIdx.x;
  for (int i = tid; i < HID * N_FEAT; i += 256) {
    int n = i >> 7, k = i & 127;
    Bs[k][n] = Wenc[i];
  }
  __syncthreads();
  int wave = tid >> 5;
  int mtile = blockIdx.x * 8 + wave;
  if (mtile >= N_NODES / 16) return;
  int lane = tid & 31, nl = lane & 15, kh = lane >> 4;
  int row = mtile * 16 + nl;
  const float* arow = x + (size_t)row * N_FEAT;
  v8f acc[4] = {};
  for (int k0 = 0; k0 < N_FEAT; k0 += 4) {
    int ka = k0 + 2 * kh;
    v2f a = *(const v2f*)(arow + ka);
#pragma unroll
    for (int t = 0; t < 4; t++) {
      v2f b;
      b.x = Bs[ka][t * 16 + nl];
      b.y = Bs[ka + 1][t * 16 + nl];
      acc[t] = wmma_f32(a, b, acc[t]);
    }
  }
  int mbase = mtile * 16 + 8 * kh;
#pragma unroll
  for (int t = 0; t < 4; t++) {
    int col = t * 16 + nl;
    float bb = benc[col];
#pragma unroll
    for (int v = 0; v < 8; v++)
      hout[(size_t)(mbase + v) * HID + col] = acc[t][v] + bb;
  }
}

// ---- dual GEMM: xp = hn@Wp^T, xn = hn@Wn^T; acc seeded with bias + self-loop ----

__global__ __launch_bounds__(256) void dual_gemm_kernel(
    const float* __restrict__ hn, const float* __restrict__ Wp, const float* __restrict__ Wn,
    const float* __restrict__ bp, const float* __restrict__ bn,
    const float* __restrict__ dinvp, const float* __restrict__ dinvn,
    float* __restrict__ xp, float* __restrict__ xn,
    float* __restrict__ accp, float* __restrict__ accn) {
  __shared__ float Bs[2][HID][HID + 1];
  int tid = threadIdx.x;
  for (int i = tid; i < HID * HID; i += 256) {
    int n = i >> 6, k = i & 63;
    Bs[0][k][n] = Wp[i];
    Bs[1][k][n] = Wn[i];
  }
  __syncthreads();
  int wave = tid >> 5;
  int mtile = blockIdx.x * 8 + wave;
  if (mtile >= N_NODES / 16) return;
  int lane = tid & 31, nl = lane & 15, kh = lane >> 4;
  int row = mtile * 16 + nl;
  const float* arow = hn + (size_t)row * HID;
  v8f ap[4] = {}, an[4] = {};
  for (int k0 = 0; k0 < HID; k0 += 4) {
    int ka = k0 + 2 * kh;
    v2f a = *(const v2f*)(arow + ka);
#pragma unroll
    for (int t = 0; t < 4; t++) {
      v2f b0, b1;
      b0.x = Bs[0][ka][t * 16 + nl];
      b0.y = Bs[0][ka + 1][t * 16 + nl];
      ap[t] = wmma_f32(a, b0, ap[t]);
      b1.x = Bs[1][ka][t * 16 + nl];
      b1.y = Bs[1][ka + 1][t * 16 + nl];
      an[t] = wmma_f32(a, b1, an[t]);
    }
  }
  int mbase = mtile * 16 + 8 * kh;
#pragma unroll
  for (int t = 0; t < 4; t++) {
    int col = t * 16 + nl;
    float bpc = bp[col], bnc = bn[col];
#pragma unroll
    for (int v = 0; v < 8; v++) {
      int r = mbase + v;
      float dp = dinvp[r], dn = dinvn[r];
      float vp = ap[t][v], vn = an[t][v];
      size_t o = (size_t)r * HID + col;
      xp[o] = vp;
      xn[o] = vn;
      accp[o] = bpc + dp * dp * vp;  // self-loop fused
      accn[o] = bnc + dn * dn * vn;
    }
  }
}

// ---------------- edge scatter: 1 wave per edge, 2 floats per lane ----------------

__global__ void edge_kernel(const long long* __restrict__ ep, const long long* __restrict__ en,
                            const float* __restrict__ dinvp, const float* __restrict__ dinvn,
                            const float* __restrict__ xp, const float* __restrict__ xn,
                            float* __restrict__ accp, float* __restrict__ accn) {
  long long tid = (long long)blockIdx.x * blockDim.x + threadIdx.x;
  long long wv = tid >> 5;
  int lane = (int)(tid & 31);
  const long long* idx;
  const float* dinv;
  const float* xs;
  float* acc;
  long long e;
  if (wv < N_EDGES) {
    idx = ep; dinv = dinvp; xs = xp; acc = accp; e = wv;
  } else {
    e = wv - N_EDGES;
    if (e >= N_EDGES) return;
    idx = en; dinv = dinvn; xs = xn; acc = accn;
  }
  int s = (int)idx[e];
  int d = (int)idx[N_EDGES + e];
  float nrm = dinv[s] * dinv[d];
  const float2 v = *(const float2*)(xs + (size_t)s * HID + 2 * lane);
  float* a = acc + (size_t)d * HID + 2 * lane;
  atomicAdd(a, nrm * v.x);
  atomicAdd(a + 1, nrm * v.y);
}

// ---- psi GEMM + RK4 stage epilog:
//   delta = clip(accp@Wpp^T + accn@Wpn^T + bpp + bpn)
//   mode 0: ksum=d;      hstage=h+0.5dt*d
//   mode 1: ksum+=2d;    hstage=h+0.5dt*d
//   mode 2: ksum+=2d;    hstage=h+dt*d
//   mode 3: h+=dt/6*(ksum+d)

__global__ __launch_bounds__(256) void psi_kernel(
    const float* __restrict__ accp, const float* __restrict__ accn,
    const float* __restrict__ Wpp, const float* __restrict__ Wpn,
    const float* __restrict__ bpp, const float* __restrict__ bpn,
    const float* __restrict__ tarr, int mode,
    float* __restrict__ h, float* __restrict__ ksum, float* __restrict__ hstage) {
  __shared__ float Bs[2][HID][HID + 1];
  int tid = threadIdx.x;
  for (int i = tid; i < HID * HID; i += 256) {
    int n = i >> 6, k = i & 63;
    Bs[0][k][n] = Wpp[i];
    Bs[1][k][n] = Wpn[i];
  }
  __syncthreads();
  int wave = tid >> 5;
  int mtile = blockIdx.x * 8 + wave;
  if (mtile >= N_NODES / 16) return;
  int lane = tid & 31, nl = lane & 15, kh = lane >> 4;
  int row = mtile * 16 + nl;
  const float* arp = accp + (size_t)row * HID;
  const float* arn = accn + (size_t)row * HID;
  v8f acc[4] = {};
  for (int k0 = 0; k0 < HID; k0 += 4) {
    int ka = k0 + 2 * kh;
    v2f a0 = *(const v2f*)(arp + ka);
    v2f a1 = *(const v2f*)(arn + ka);
#pragma unroll
    for (int t = 0; t < 4; t++) {
      v2f b0, b1;
      b0.x = Bs[0][ka][t * 16 + nl];
      b0.y = Bs[0][ka + 1][t * 16 + nl];
      acc[t] = wmma_f32(a0, b0, acc[t]);
      b1.x = Bs[1][ka][t * 16 + nl];
      b1.y = Bs[1][ka + 1][t * 16 + nl];
      acc[t] = wmma_f32(a1, b1, acc[t]);
    }
  }
  float dt = (tarr[1] - tarr[0]) * 0.5f;  // (t1-t0)/RK4_STEPS
  int mbase = mtile * 16 + 8 * kh;
#pragma unroll
  for (int t = 0; t < 4; t++) {
    int col = t * 16 + nl;
    float bc = bpp[col] + bpn[col];
#pragma unroll
    for (int v = 0; v < 8; v++) {
      size_t o = (size_t)(mbase + v) * HID + col;
      float d = acc[t][v] + bc;
      d = fminf(fmaxf(d, -50.0f), 50.0f);
      if (mode == 0) {
        ksum[o] = d;
        hstage[o] = h[o] + 0.5f * dt * d;
      } else if (mode == 1) {
        ksum[o] += 2.0f * d;
        hstage[o] = h[o] + 0.5f * dt * d;
      } else if (mode == 2) {
        ksum[o] += 2.0f * d;
        hstage[o] = h[o] + dt * d;
      } else {
        h[o] = h[o] + (dt * (1.0f / 6.0f)) * (ksum[o] + d);
      }
    }
  }
}

// ---------------- host ----------------

extern "C" void kernel_launch(void* const* d_in, const int* in_sizes, int n_in,
                              void* d_out, int out_size, void* d_ws, size_t ws_size,
                              hipStream_t stream) {
  const float* x = (const float*)d_in[0];
  const long long* ep = (const long long*)d_in[1];
  const long long* en = (const long long*)d_in[2];
  const float* t = (const float*)d_in[3];
  const float* Wenc = (const float*)d_in[4];
  const float* benc = (const float*)d_in[5];
  const float* Wp = (const float*)d_in[6];
  const float* bp = (const float*)d_in[7];
  const float* Wn = (const float*)d_in[8];
  const float* bn = (const float*)d_in[9];
  const float* Wpp = (const float*)d_in[10];
  const float* bpp = (const float*)d_in[11];
  const float* Wpn = (const float*)d_in[12];
  const float* bpn = (const float*)d_in[13];
  const float* gamma = (const float*)d_in[14];
  const float* beta = (const float*)d_in[15];

  float* h = (float*)d_out;
  float* ws = (float*)d_ws;
  const size_t NH = (size_t)N_NODES * HID;
  float* hn = ws;
  float* xp = hn + NH;
  float* xn = xp + NH;
  float* accp = xn + NH;
  float* accn = accp + NH;
  float* ksum = accn + NH;
  float* hstage = ksum + NH;
  float* dinvp = hstage + NH;
  float* dinvn = dinvp + N_NODES;  // contiguous with dinvp

  // degrees (self-loop => init 1.0), then rsqrt in place
  fill_kernel<<<(2 * N_NODES + 255) / 256, 256, 0, stream>>>(dinvp, 2 * N_NODES, 1.0f);
  deg_kernel<<<(2 * N_EDGES + 255) / 256, 256, 0, stream>>>(ep, en, dinvp, dinvn);
  dinv_kernel<<<(N_NODES + 255) / 256, 256, 0, stream>>>(dinvp, dinvn);

  const int mtiles = N_NODES / 16;        // 3125 (exact)
  const int gblocks = (mtiles + 7) / 8;   // 391
  enc_gemm_kernel<<<gblocks, 256, 0, stream>>>(x, Wenc, benc, h);

  const int ln_blocks = (N_NODES * 32 + 255) / 256;
  const int edge_blocks = (int)((2LL * N_EDGES * 32 + 255) / 256);

  for (int step = 0; step < 2; step++) {
    for (int sub = 0; sub < 4; sub++) {
      const float* fin = (sub == 0) ? h : hstage;
      ln_kernel<<<ln_blocks, 256, 0, stream>>>(fin, hn, gamma, beta);
      dual_gemm_kernel<<<gblocks, 256, 0, stream>>>(hn, Wp, Wn, bp, bn, dinvp, dinvn,
                                                    xp, xn, accp, accn);
      edge_kernel<<<edge_blocks, 256, 0, stream>>>(ep, en, dinvp, dinvn, xp, xn, accp, accn);
      psi_kernel<<<gblocks, 256, 0, stream>>>(accp, accn, Wpp, Wpn, bpp, bpn, t, sub,
                                              h, ksum, hstage);
    }
  }
}